// ComplexGaussianRasterizer_24627342475320
// MI455X (gfx1250) — compile-verified
//
#include <hip/hip_runtime.h>
#include <hip/hip_bf16.h>
#include <stdint.h>

// ---------------- problem constants (from reference) ----------------
#define NRES      128
#define KWIN      6            // K
#define CELLS     216          // K^3
#define REC_DW    16           // dwords per per-gaussian record
#define GPB       16           // gaussians per splat block
#define SPLAT_TPB 256          // 8 wave32s

// 0.5 * log2(e): exp(-0.5*q) == exp2(-S*q)
#define QSCALE 0.72134752044448170f

typedef unsigned int u32x4 __attribute__((ext_vector_type(4)));
typedef int          i32x8 __attribute__((ext_vector_type(8)));
typedef int          i32x4 __attribute__((ext_vector_type(4)));

#if __has_builtin(__builtin_amdgcn_tensor_load_to_lds) && __has_builtin(__builtin_amdgcn_s_wait_tensorcnt)
#define HAVE_TDM 1
#else
#define HAVE_TDM 0
#endif

// ---------------- per-gaussian preprocessing ----------------
struct GRec {
    float a0,a1,a2,a3,a4,a5,a6,a7,a8,a9;  // folded quadratic-form coeffs (pre-scaled)
    float cr, ci;                          // opacity*cos(ph), opacity*(sin(ph)+ph_add)
    int   bx, by, bz;                      // base voxel
};

__device__ __forceinline__ GRec make_record(int n,
        const float* __restrict__ means, const float* __restrict__ opac,
        const float* __restrict__ scales, const float* __restrict__ rots,
        const float* __restrict__ phases, const float* __restrict__ phadd)
{
    const float lb  = -1.0f;
    const float vox = 2.0f / (float)NRES;   // 0.015625

    float mx = means[3*n+0], my = means[3*n+1], mz = means[3*n+2];
    float sx = scales[3*n+0], sy = scales[3*n+1], sz = scales[3*n+2];
    float qw = rots[4*n+0], qx = rots[4*n+1], qy = rots[4*n+2], qz = rots[4*n+3];

    float rn = rsqrtf(qw*qw + qx*qx + qy*qy + qz*qz);
    qw *= rn; qx *= rn; qy *= rn; qz *= rn;

    // rotation matrix
    float r00 = 1.f - 2.f*(qy*qy + qz*qz), r01 = 2.f*(qx*qy - qw*qz), r02 = 2.f*(qx*qz + qw*qy);
    float r10 = 2.f*(qx*qy + qw*qz), r11 = 1.f - 2.f*(qx*qx + qz*qz), r12 = 2.f*(qy*qz - qw*qx);
    float r20 = 2.f*(qx*qz - qw*qy), r21 = 2.f*(qy*qz + qw*qx), r22 = 1.f - 2.f*(qx*qx + qy*qy);

    // M = R * diag(s);  cov = M M^T (symmetric)
    float m00=r00*sx, m01=r01*sy, m02=r02*sz;
    float m10=r10*sx, m11=r11*sy, m12=r12*sz;
    float m20=r20*sx, m21=r21*sy, m22=r22*sz;
    float c00 = m00*m00 + m01*m01 + m02*m02;
    float c01 = m00*m10 + m01*m11 + m02*m12;
    float c02 = m00*m20 + m01*m21 + m02*m22;
    float c11 = m10*m10 + m11*m11 + m12*m12;
    float c12 = m10*m20 + m11*m21 + m12*m22;
    float c22 = m20*m20 + m21*m21 + m22*m22;

    // symmetric 3x3 inverse via adjugate
    float k00 = c11*c22 - c12*c12;
    float k01 = c02*c12 - c01*c22;
    float k02 = c01*c12 - c02*c11;
    float det = c00*k00 + c01*k01 + c02*k02;
    float id  = 1.0f / det;
    float a00 = k00*id, a01 = k01*id, a02 = k02*id;
    float a11 = (c00*c22 - c02*c02)*id;
    float a12 = (c02*c01 - c00*c12)*id;
    float a22 = (c00*c11 - c01*c01)*id;

    // base voxel and displacement of cell (0,0,0) center
    int bx = (int)floorf((mx - lb) / vox) - (KWIN/2);
    int by = (int)floorf((my - lb) / vox) - (KWIN/2);
    int bz = (int)floorf((mz - lb) / vox) - (KWIN/2);
    float dx = lb + ((float)bx + 0.5f)*vox - mx;
    float dy = lb + ((float)by + 0.5f)*vox - my;
    float dz = lb + ((float)bz + 0.5f)*vox - mz;

    // t = A * d0
    float tx = a00*dx + a01*dy + a02*dz;
    float ty = a01*dx + a11*dy + a12*dz;
    float tz = a02*dx + a12*dy + a22*dz;

    // quad(i,j,k)*QSCALE = a0 + i*a1 + j*a2 + k*a3 + i*i*a4 + i*j*a5 + i*k*a6 + j*j*a7 + j*k*a8 + k*k*a9
    const float S  = QSCALE;
    const float Sv = S*vox, Sv2 = S*vox*vox;
    GRec r;
    r.a0 = S*(dx*tx + dy*ty + dz*tz);
    r.a1 = 2.f*Sv*tx;   r.a2 = 2.f*Sv*ty;   r.a3 = 2.f*Sv*tz;
    r.a4 = Sv2*a00;     r.a5 = 2.f*Sv2*a01; r.a6 = 2.f*Sv2*a02;
    r.a7 = Sv2*a11;     r.a8 = 2.f*Sv2*a12; r.a9 = Sv2*a22;

    float op = opac[n], ph = phases[n], pa = phadd[n];
    float sp, cp;
    __sincosf(ph, &sp, &cp);
    r.cr = op * cp;
    r.ci = op * (sp + pa);
    r.bx = bx; r.by = by; r.bz = bz;
    return r;
}

// ---------------- kernels ----------------
__global__ void zero_kernel(float* __restrict__ out, int n)
{
    int n4  = n >> 2;
    int idx = blockIdx.x * blockDim.x + threadIdx.x;
    float4 z = make_float4(0.f, 0.f, 0.f, 0.f);
    for (int i = idx; i < n4; i += gridDim.x * blockDim.x)
        ((float4*)out)[i] = z;
    if (idx == 0)
        for (int i = n4 << 2; i < n; ++i) out[i] = 0.f;
}

__global__ void prep_kernel(const float* __restrict__ means, const float* __restrict__ opac,
                            const float* __restrict__ scales, const float* __restrict__ rots,
                            const float* __restrict__ phases, const float* __restrict__ phadd,
                            float* __restrict__ ws, int N)
{
    int n = blockIdx.x * blockDim.x + threadIdx.x;
    if (n >= N) return;
    GRec r = make_record(n, means, opac, scales, rots, phases, phadd);
    float* o = ws + (size_t)n * REC_DW;
    ((float4*)o)[0] = make_float4(r.a0, r.a1, r.a2, r.a3);
    ((float4*)o)[1] = make_float4(r.a4, r.a5, r.a6, r.a7);
    ((float4*)o)[2] = make_float4(r.a8, r.a9, r.cr, r.ci);
    ((float4*)o)[3] = make_float4(__int_as_float(r.bx), __int_as_float(r.by),
                                  __int_as_float(r.bz), 0.f);
}

__device__ __forceinline__ void splat_cell(const float* __restrict__ R, int i, int j, int k,
                                           float* __restrict__ out)
{
    int vx = __float_as_int(R[12]) + i;
    int vy = __float_as_int(R[13]) + j;
    int vz = __float_as_int(R[14]) + k;
    if ((unsigned)vx >= (unsigned)NRES || (unsigned)vy >= (unsigned)NRES ||
        (unsigned)vz >= (unsigned)NRES)
        return;                      // reference adds exactly 0 here
    float fi = (float)i, fj = (float)j, fk = (float)k;
    float q = R[0];
    q = fmaf(fi,    R[1], q);
    q = fmaf(fj,    R[2], q);
    q = fmaf(fk,    R[3], q);
    q = fmaf(fi*fi, R[4], q);
    q = fmaf(fi*fj, R[5], q);
    q = fmaf(fi*fk, R[6], q);
    q = fmaf(fj*fj, R[7], q);
    q = fmaf(fj*fk, R[8], q);
    q = fmaf(fk*fk, R[9], q);
#if __has_builtin(__builtin_amdgcn_exp2f)
    float w = __builtin_amdgcn_exp2f(-q);   // exp(-0.5*quad), scale pre-folded
#else
    float w = exp2f(-q);
#endif
    int flat = (((vx * NRES) + vy) * NRES + vz) * 2;
    unsafeAtomicAdd(&out[flat],     w * R[10]);   // native global_atomic_add_f32
    unsafeAtomicAdd(&out[flat + 1], w * R[11]);
}

__global__ void __launch_bounds__(SPLAT_TPB)
splat_kernel(const float* __restrict__ recs, float* __restrict__ out, int N)
{
    __shared__ __align__(16) float smem[GPB * REC_DW];
    int blockStart = blockIdx.x * GPB;
    int nG = N - blockStart;
    if (nG > GPB) nG = GPB;
    unsigned words = (unsigned)nG * REC_DW;          // dwords to stage (<= 256)

#if HAVE_TDM
    if (threadIdx.x < 32) {                           // wave 0 issues the TDM DMA
        uint32_t ldsAddr = (uint32_t)(uintptr_t)(void*)smem;            // LDS byte offset
        uint64_t gaddr   = (uint64_t)(uintptr_t)(recs + (size_t)blockStart * REC_DW);
        // D# group0: count=1 | lds_addr | global_addr | type=2
        u32x4 g0;
        g0[0] = 1u;
        g0[1] = ldsAddr;
        g0[2] = (uint32_t)gaddr;
        g0[3] = ((uint32_t)(gaddr >> 32) & 0x01FFFFFFu) | (2u << 30);
        // D# group1: data_size=4B, 1-D tile of `words` elements
        i32x8 g1;
        g1[0] = (int)(2u << 16);                      // data_size = 2 (4 bytes)
        g1[1] = (int)((words & 0xFFFFu) << 16);       // tensor_dim0[15:0]
        g1[2] = (int)((words >> 16) | (1u << 16));    // tensor_dim0[31:16] | tensor_dim1=1
        g1[3] = (int)(words << 16);                   // tile_dim0 = words
        g1[4] = 0;                                    // tile_dim1 = tile_dim2 = 0
        g1[5] = (int)words;                           // tensor_dim0_stride
        g1[6] = 0;
        g1[7] = 0;
        i32x4 gz4 = (i32x4)0;                         // groups 2/3 unused (1-D tile)
        i32x8 gz8 = (i32x8)0;                         // extra group in 6-arg form
        __builtin_amdgcn_tensor_load_to_lds(g0, g1, gz4, gz4, gz8, 0);
        __builtin_amdgcn_s_wait_tensorcnt(0);
    }
#else
    for (int t = threadIdx.x; t < (int)words; t += blockDim.x)
        smem[t] = recs[(size_t)blockStart * REC_DW + t];
#endif
    __syncthreads();

    int total = nG * CELLS;
    for (int idx = threadIdx.x; idx < total; idx += SPLAT_TPB) {
        int g   = idx / CELLS;
        int c   = idx - g * CELLS;
        int i   = c / 36;
        int rem = c - i * 36;
        int j   = rem / 6;
        int k   = rem - j * 6;
        splat_cell(&smem[g * REC_DW], i, j, k, out);
    }
}

// fallback if workspace is too small: fully fused, one thread per gaussian
__global__ void fused_kernel(const float* __restrict__ means, const float* __restrict__ opac,
                             const float* __restrict__ scales, const float* __restrict__ rots,
                             const float* __restrict__ phases, const float* __restrict__ phadd,
                             float* __restrict__ out, int N)
{
    int n = blockIdx.x * blockDim.x + threadIdx.x;
    if (n >= N) return;
    GRec r = make_record(n, means, opac, scales, rots, phases, phadd);
    float R[16];
    R[0]=r.a0; R[1]=r.a1; R[2]=r.a2; R[3]=r.a3; R[4]=r.a4; R[5]=r.a5;
    R[6]=r.a6; R[7]=r.a7; R[8]=r.a8; R[9]=r.a9; R[10]=r.cr; R[11]=r.ci;
    R[12]=__int_as_float(r.bx); R[13]=__int_as_float(r.by); R[14]=__int_as_float(r.bz);
    for (int i = 0; i < KWIN; ++i)
        for (int j = 0; j < KWIN; ++j)
            for (int k = 0; k < KWIN; ++k)
                splat_cell(R, i, j, k, out);
}

// ---------------- host ----------------
extern "C" void kernel_launch(void* const* d_in, const int* in_sizes, int n_in,
                              void* d_out, int out_size, void* d_ws, size_t ws_size,
                              hipStream_t stream)
{
    const float* means  = (const float*)d_in[0];
    const float* opac   = (const float*)d_in[1];
    const float* scales = (const float*)d_in[2];
    const float* rots   = (const float*)d_in[3];
    const float* phases = (const float*)d_in[4];
    const float* phadd  = (const float*)d_in[5];
    float* out = (float*)d_out;
    int N = in_sizes[1];                      // opacities: one per gaussian

    // clear the accumulation grid (16 MB, L2-resident)
    {
        int n4 = out_size >> 2;
        int blocks = (n4 + 255) / 256;
        if (blocks < 1) blocks = 1;
        if (blocks > 8192) blocks = 8192;
        zero_kernel<<<blocks, 256, 0, stream>>>(out, out_size);
    }

    size_t need = (size_t)N * REC_DW * sizeof(float);
    if (ws_size >= need) {
        prep_kernel<<<(N + 255) / 256, 256, 0, stream>>>(
            means, opac, scales, rots, phases, phadd, (float*)d_ws, N);
        int blocks = (N + GPB - 1) / GPB;
        splat_kernel<<<blocks, SPLAT_TPB, 0, stream>>>((const float*)d_ws, out, N);
    } else {
        fused_kernel<<<(N + 255) / 256, 256, 0, stream>>>(
            means, opac, scales, rots, phases, phadd, out, N);
    }
}